// Encoder_38869454028868
// MI455X (gfx1250) — compile-verified
//
#include <hip/hip_runtime.h>

// WaveNet res-stack for MI455X (gfx1250): 24 x [dilated conv1d(k=2) + bias + SiLU + residual]
// Each layer = two 128x128 f32 GEMMs on time-shifted x  ->  V_WMMA_F32_16X16X4_F32.
// x (128 MB) ping-pongs between d_ws and d_out and stays resident in the 192 MB L2.
// Intermediate x buffers are TIME-MAJOR [t][c]: LDS fills are contiguous b128 async
// copies (ASYNCcnt path) and epilogue stores are contiguous b128. Only layer 0
// (reads channel-major `a`) transposes on fill; only layer 23 scatters on store.
// K loop is fully unrolled with lane-variant offsets folded into base pointers so
// every ds/global access in the hot loop uses immediate offsets (no VALU address math).

typedef float v2f __attribute__((ext_vector_type(2)));
typedef float v4f __attribute__((ext_vector_type(4)));
typedef float v8f __attribute__((ext_vector_type(8)));
typedef int   v4i __attribute__((ext_vector_type(4)));

#define C_CH    128
#define L_LEN   32768
#define B_SZ    8
#define TILE_T  128
#define HALO    16                    // covers max shift 2^(5-1) = 16
#define TPAD    (TILE_T + 2 * HALO)   // 160 time rows in LDS (1.25x read amp)
#define CSTRIDE 132                   // 128 ch + 4 pad -> 16B-aligned rows, bank-clean
#define NT      (TILE_T / 16)         // 8 N-tiles per block

#if __has_builtin(__builtin_amdgcn_global_load_async_to_lds_b32)
#define HAVE_ASYNC32 1
typedef __attribute__((address_space(1))) int async_gint;   // global int
typedef __attribute__((address_space(3))) int async_lint;   // LDS int
#else
#define HAVE_ASYNC32 0
#endif

#if __has_builtin(__builtin_amdgcn_global_load_async_to_lds_b128)
#define HAVE_ASYNC128 1
typedef __attribute__((address_space(1))) v4i async_gv4i;   // global int4
typedef __attribute__((address_space(3))) v4i async_lv4i;   // LDS int4
#else
#define HAVE_ASYNC128 0
#endif

__device__ __forceinline__ void wait_async_then_barrier() {
#if HAVE_ASYNC32 || HAVE_ASYNC128
#if __has_builtin(__builtin_amdgcn_s_wait_asynccnt)
    __builtin_amdgcn_s_wait_asynccnt(0);
#else
    asm volatile("s_wait_asynccnt 0x0" ::: "memory");
#endif
#endif
    __syncthreads();
}

template <bool SRC_CMAJOR, bool DST_CMAJOR>
__global__ __launch_bounds__(256) void wavenet_layer_kernel(
    const float* __restrict__ src,   // x input:  [B][C][L] if SRC_CMAJOR else [B][L][C]
    const float* __restrict__ Wl,    // [128][128][2] this layer's weights
    const float* __restrict__ bl,    // [128] this layer's bias
    float* __restrict__ dst,         // x output: [B][C][L] if DST_CMAJOR else [B][L][C]
    int off0, int off1)              // tap time offsets (tap0 <= 0 <= tap1)
{
    __shared__ float xs[TPAD * CSTRIDE];   // time-major tile: xs[t][c], ~83 KB

    const int tid  = threadIdx.x;
    const int lane = tid & 31;
    const int wid  = tid >> 5;              // 0..7 -> M tile (16 out-channels)
    const int bi   = blockIdx.y;
    const int t0   = blockIdx.x * TILE_T;

    // warm L2/L0 with this layer's 128 KB weight slice (global_prefetch_b8)
    __builtin_prefetch((const char*)Wl + tid * 512, 0, 3);

    // ---- cooperative LDS fill of x tile (+halo), zero padding outside [0, L) ----
    const float* xb = src + (size_t)bi * C_CH * L_LEN;
    if (SRC_CMAJOR) {
        // transpose fill: global [c][t] -> LDS [t][c], per-element b32
        for (int idx = tid; idx < C_CH * TPAD; idx += 256) {
            int c  = idx / TPAD;
            int tt = idx - c * TPAD;
            int gt = t0 - HALO + tt;
#if HAVE_ASYNC32
            if (gt >= 0 && gt < L_LEN) {
                __builtin_amdgcn_global_load_async_to_lds_b32(
                    (async_gint*)(xb + (size_t)c * L_LEN + gt),
                    (async_lint*)&xs[tt * CSTRIDE + c], 0, 0);
            } else {
                xs[tt * CSTRIDE + c] = 0.0f;
            }
#else
            float v = 0.0f;
            if (gt >= 0 && gt < L_LEN) v = xb[(size_t)c * L_LEN + gt];
            xs[tt * CSTRIDE + c] = v;
#endif
        }
    } else {
        // straight row copy: global [t][c] -> LDS [t][c], 16B per lane per op
        for (int idx = tid; idx < TPAD * (C_CH / 4); idx += 256) {
            int row = idx >> 5;             // time row (C_CH/4 == 32 quads per row)
            int q   = idx & 31;
            int gt  = t0 - HALO + row;
            float* ld = &xs[row * CSTRIDE + q * 4];
            if (gt >= 0 && gt < L_LEN) {
#if HAVE_ASYNC128
                __builtin_amdgcn_global_load_async_to_lds_b128(
                    (async_gv4i*)(xb + (size_t)gt * C_CH + q * 4),
                    (async_lv4i*)ld, 0, 0);
#else
                *(v4f*)ld = *(const v4f*)(xb + (size_t)gt * C_CH + q * 4);
#endif
            } else {
                v4f z = {};
                *(v4f*)ld = z;
            }
        }
    }
    wait_async_then_barrier();

    const int ml    = lane & 15;            // row-in-tile (A) / col-in-tile (B,D)
    const int kr    = (lane >> 4) << 1;     // K sub-row: 0 or 2
    const int ohalf = (lane >> 4) << 3;     // D rows: lanes 16..31 hold M=r+8

    // bias values for the 8 D rows this lane contributes to (N-independent)
    float bv[8];
#pragma unroll
    for (int r = 0; r < 8; ++r) bv[r] = bl[16 * wid + r + ohalf];

    v8f zero = {};
    v8f acc[NT];
#pragma unroll
    for (int n = 0; n < NT; ++n) acc[n] = zero;

    // per-(N-tile, tap) LDS bases with the lane-variant K sub-row folded in:
    // hot-loop address = base + k0 (k0 compile-time under full unroll)
    int tbase0[NT], tbase1[NT];
#pragma unroll
    for (int n = 0; n < NT; ++n) {
        tbase0[n] = (n * 16 + ml + off0 + HALO) * CSTRIDE + kr;
        tbase1[n] = (n * 16 + ml + off1 + HALO) * CSTRIDE + kr;
    }

    const int oa = 16 * wid + ml;           // A-fragment out-channel row
    const float* wp = Wl + ((size_t)oa * C_CH + kr) * 2;   // kr folded in

    // ---- K loop, fully unrolled: pure immediate-offset ds_load_b64 + WMMA ----
#pragma unroll
    for (int k0 = 0; k0 < C_CH; k0 += 4) {
        // W[o][k][0..1], W[o][k+1][0..1] : one 16B load (imm offset) covers both taps
        v4f wv = *(const v4f*)(wp + k0 * 2);
        v2f a0; a0.x = wv.x; a0.y = wv.z;   // tap0 weights for K=k, k+1
        v2f a1; a1.x = wv.y; a1.y = wv.w;   // tap1 weights
#pragma unroll
        for (int n = 0; n < NT; ++n) {
            // time-major tile: {xs[t][k], xs[t][k+1]} contiguous, 8B aligned
            v2f b0 = *(const v2f*)(&xs[tbase0[n] + k0]);
            v2f b1 = *(const v2f*)(&xs[tbase1[n] + k0]);
            acc[n] = __builtin_amdgcn_wmma_f32_16x16x4_f32(
                false, a0, false, b0, (short)0, acc[n], false, false);
            acc[n] = __builtin_amdgcn_wmma_f32_16x16x4_f32(
                false, a1, false, b1, (short)0, acc[n], false, false);
        }
    }

    // ---- epilogue: bias + SiLU + residual (residual read from same LDS tile) ----
    float* db = dst + (size_t)bi * C_CH * L_LEN;
    const int o0 = 16 * wid + ohalf;        // first of this lane's 8 channels
#pragma unroll
    for (int n = 0; n < NT; ++n) {
        int tl = n * 16 + ml;
        if (DST_CMAJOR) {
            // scattered per-channel stores into [c][t]
#pragma unroll
            for (int r = 0; r < 8; ++r) {
                float h = acc[n][r] + bv[r];
                float s = h / (1.0f + __expf(-h));      // SiLU
                float xin = xs[(HALO + tl) * CSTRIDE + o0 + r];
                db[(size_t)(o0 + r) * L_LEN + (t0 + tl)] = xin + s;
            }
        } else {
            // contiguous 8 channels at one t: 2x v4f LDS read + 2x b128 store
            v4f x0 = *(const v4f*)(&xs[(HALO + tl) * CSTRIDE + o0]);
            v4f x1 = *(const v4f*)(&xs[(HALO + tl) * CSTRIDE + o0 + 4]);
            v4f y0, y1;
#pragma unroll
            for (int r = 0; r < 4; ++r) {
                float h0 = acc[n][r] + bv[r];
                float h1 = acc[n][r + 4] + bv[r + 4];
                y0[r] = x0[r] + h0 / (1.0f + __expf(-h0));
                y1[r] = x1[r] + h1 / (1.0f + __expf(-h1));
            }
            float* dp = db + (size_t)(t0 + tl) * C_CH + o0;
            *(v4f*)dp = y0;
            *(v4f*)(dp + 4) = y1;
        }
    }
}

extern "C" void kernel_launch(void* const* d_in, const int* in_sizes, int n_in,
                              void* d_out, int out_size, void* d_ws, size_t ws_size,
                              hipStream_t stream) {
    const float* a = (const float*)d_in[0];   // [8][128][32768] channel-major
    const float* W = (const float*)d_in[1];   // [24][128][128][2]
    const float* b = (const float*)d_in[2];   // [24][128]
    float* out = (float*)d_out;
    float* ws  = (float*)d_ws;                // needs >= 128 MB (one x-sized buffer)

    dim3 grid(L_LEN / TILE_T, B_SZ);
    dim3 block(256);

    // Ping-pong: even layer -> ws, odd layer -> out (time-major intermediates;
    // d_out's bytes are reused as scratch until layer 23 writes the final
    // channel-major result). Layer 23 (odd) ends in d_out.
    for (int layer = 0; layer < 24; ++layer) {
        int d    = layer % 6;
        int off0 = (d == 0) ? -1 : -(1 << (d - 1));
        int off1 = (d == 0) ?  0 :  (1 << (d - 1));
        const float* src = (layer == 0) ? a : ((layer & 1) ? ws : out);
        float*       dst = (layer & 1) ? out : ws;
        const float* Wp = W + (size_t)layer * C_CH * C_CH * 2;
        const float* bp = b + (size_t)layer * C_CH;
        if (layer == 0) {
            wavenet_layer_kernel<true, false><<<grid, block, 0, stream>>>(
                src, Wp, bp, dst, off0, off1);
        } else if (layer == 23) {
            wavenet_layer_kernel<false, true><<<grid, block, 0, stream>>>(
                src, Wp, bp, dst, off0, off1);
        } else {
            wavenet_layer_kernel<false, false><<<grid, block, 0, stream>>>(
                src, Wp, bp, dst, off0, off1);
        }
    }
}